// GNN_base_19748259627553
// MI455X (gfx1250) — compile-verified
//
#include <hip/hip_runtime.h>

// ---------------------------------------------------------------------------
// Fused GNN forward for MI455X (gfx1250, wave32).
//   enc    : per-sample 16x32 @ 32x64 (+bias, relu)   -> v_wmma_f32_16x16x4_f32
//   pool   : validity-masked mean over enc rows 1..15 (ballot/ctz + shfl)
//   hidden : 16x128 @ 128x128 (+bias, relu)           -> v_wmma_f32_16x16x4_f32
//   logits : 16x128 @ 128x8   (+bias)                 -> v_wmma_f32_16x16x4_f32
// fp32 WMMA keeps exact reference numerics; kernel is HBM-bound (~65MB total),
// weights stay resident in 192MB L2 so B-fragments are read from global.
// ---------------------------------------------------------------------------

typedef float v2f __attribute__((ext_vector_type(2)));
typedef float v8f __attribute__((ext_vector_type(8)));

#define N_STATES   16
#define STATE_SIZE 32
#define ENC        64
#define HID        128
#define OUT_DIM    8
#define TILE       16       // samples per block
#define BLOCK_THREADS 256   // 8 waves (wave32)

__global__ __launch_bounds__(BLOCK_THREADS)
void gnn_fused_wmma(const float* __restrict__ obs,
                    const float* __restrict__ W_enc, const float* __restrict__ b_enc,
                    const float* __restrict__ W_f,   const float* __restrict__ b_f,
                    const float* __restrict__ W_dec, const float* __restrict__ b_dec,
                    float* __restrict__ out, int n_samples)
{
    __shared__ float f_in[TILE * 2 * ENC];   // 16 x 128 concat(agent, agg)
    __shared__ float hidden[TILE * HID];     // 16 x 128

    const int tid   = threadIdx.x;
    const int wave  = tid >> 5;
    const int lane  = tid & 31;
    const int lo    = lane & 15;     // column / row index within 16
    const int hi    = lane >> 4;     // half-wave select
    const int khalf = hi * 2;        // K offset for A/B fragment layout
    const int mbase = hi * 8;        // row base for C/D fragment layout
    const int blockBase = blockIdx.x * TILE;

    // ---------------- Phase 1: encoder GEMM + masked mean-pool ----------------
    for (int s = 0; s < 2; ++s) {
        const int ls = wave * 2 + s;                         // local sample 0..15
        const int b  = blockBase + ls;                       // global sample
        const float* st = obs + (size_t)b * (N_STATES * STATE_SIZE);
        if (s == 0) __builtin_prefetch(st + N_STATES * STATE_SIZE, 0, 0);

        // A fragments: A[M=16 rows (states)][K=32 feats], f32 16x16x4 layout:
        //   lanes 0-15 : M=lane,    K = 4k+0, 4k+1   (contiguous -> b64 load)
        //   lanes 16-31: M=lane-16, K = 4k+2, 4k+3   (contiguous -> b64 load)
        v2f a[8];
#pragma unroll
        for (int k = 0; k < 8; ++k)
            a[k] = *(const v2f*)(st + lo * STATE_SIZE + k * 4 + khalf);

        // validity cumprod: flags[n] = states[n][0] lives in a[0].x on lanes 0-15
        const float flag = a[0][0];
        const bool ok = (hi == 0) ? (flag == 1.0f) : true;
        const unsigned msk   = __builtin_amdgcn_ballot_w32(ok);
        const unsigned inval = ((~msk) >> 1) & 0x7FFFu;      // bits for n=1..15
        const int   cnt = __builtin_ctz(inval | 0x8000u);    // leading-valid count 0..15
        const float inv = 1.0f / (float)(cnt > 1 ? cnt : 1);

#pragma unroll
        for (int t = 0; t < 4; ++t) {                        // N-tiles of the 16x64 enc
            const int n = lo + t * 16;
            v8f acc = {};
#pragma unroll
            for (int k = 0; k < 8; ++k) {                    // K = 32 in steps of 4
                const int kk = k * 4 + khalf;
                v2f bf;
                bf[0] = W_enc[kk * ENC + n];
                bf[1] = W_enc[(kk + 1) * ENC + n];
                acc = __builtin_amdgcn_wmma_f32_16x16x4_f32(
                        false, a[k], false, bf, (short)0, acc, false, false);
            }
            // C/D layout: acc[v] = enc[M = v+mbase][N = n]
            const float bias = b_enc[n];
            float part = 0.0f, agent = 0.0f;
#pragma unroll
            for (int v = 0; v < 8; ++v) {
                const int m = v + mbase;
                float e = acc[v] + bias;
                e = e > 0.0f ? e : 0.0f;                     // relu
                if (m == 0) agent = e;                       // agent row
                const float w = (m >= 1 && m <= cnt) ? inv : 0.0f;
                part += w * e;                               // masked mean partial
            }
            part += __shfl_xor(part, 16, 32);                // add other half's rows
            if (hi == 0) {
                f_in[ls * (2 * ENC) + n]       = agent;
                f_in[ls * (2 * ENC) + ENC + n] = part;
            }
        }
    }
    __syncthreads();

    // ---------------- Phase 2: hidden = relu(f_in(16x128) @ W_f(128x128) + b_f)
    {
        const int n = lo + wave * 16;                        // each wave: one N-tile
        v8f acc = {};
#pragma unroll 4
        for (int k = 0; k < 32; ++k) {                       // K = 128
            const int kk = k * 4 + khalf;
            v2f av = *(const v2f*)&f_in[lo * (2 * ENC) + kk];
            v2f bf;
            bf[0] = W_f[kk * HID + n];
            bf[1] = W_f[(kk + 1) * HID + n];
            acc = __builtin_amdgcn_wmma_f32_16x16x4_f32(
                    false, av, false, bf, (short)0, acc, false, false);
        }
        const float bias = b_f[n];
#pragma unroll
        for (int v = 0; v < 8; ++v) {
            float e = acc[v] + bias;
            hidden[(v + mbase) * HID + n] = e > 0.0f ? e : 0.0f;
        }
    }
    __syncthreads();

    // ---------------- Phase 3: logits = hidden(16x128) @ W_dec(128x8) + b_dec
    if (wave == 0) {                                         // wave-uniform: EXEC stays full
        const int n = lo;
        const float nmask = (n < OUT_DIM) ? 1.0f : 0.0f;     // zero-pad N to 16 w/o divergence
        const int nc = n & (OUT_DIM - 1);
        v8f acc = {};
#pragma unroll 4
        for (int k = 0; k < 32; ++k) {
            const int kk = k * 4 + khalf;
            v2f av = *(const v2f*)&hidden[lo * HID + kk];
            v2f bf;
            bf[0] = W_dec[kk * OUT_DIM + nc] * nmask;
            bf[1] = W_dec[(kk + 1) * OUT_DIM + nc] * nmask;
            acc = __builtin_amdgcn_wmma_f32_16x16x4_f32(
                    false, av, false, bf, (short)0, acc, false, false);
        }
        if (n < OUT_DIM) {
            const float bias = b_dec[n];
#pragma unroll
            for (int v = 0; v < 8; ++v) {
                const int m = v + mbase;
                out[(size_t)(blockBase + m) * OUT_DIM + n] = acc[v] + bias;
            }
        }
    }
}

extern "C" void kernel_launch(void* const* d_in, const int* in_sizes, int n_in,
                              void* d_out, int out_size, void* d_ws, size_t ws_size,
                              hipStream_t stream) {
    (void)n_in; (void)out_size; (void)d_ws; (void)ws_size;
    const float* obs   = (const float*)d_in[0];
    const float* W_enc = (const float*)d_in[1];
    const float* b_enc = (const float*)d_in[2];
    const float* W_f   = (const float*)d_in[3];
    const float* b_f   = (const float*)d_in[4];
    const float* W_dec = (const float*)d_in[5];
    const float* b_dec = (const float*)d_in[6];
    float* out = (float*)d_out;

    const int n_samples = in_sizes[0] / (N_STATES * STATE_SIZE);
    if (n_samples <= 0) return;
    const int grid = n_samples / TILE;   // B=32768 -> 2048 blocks of 16 samples

    gnn_fused_wmma<<<grid, BLOCK_THREADS, 0, stream>>>(
        obs, W_enc, b_enc, W_f, b_f, W_dec, b_dec, out, n_samples);
}